// JKNet_53154515255820
// MI455X (gfx1250) — compile-verified
//
#include <hip/hip_runtime.h>

#define NN 100000
#define NE 1600000
#define HD 128
#define NL 4
#define NG 256
#define ENCD 256
#define DECD 128
#define BN_EPS 1e-5f

typedef __attribute__((ext_vector_type(2))) float v2f;
typedef __attribute__((ext_vector_type(8))) float v8f;

// ---------------------------------------------------------------- utilities
__global__ void fill_kernel(float* __restrict__ p, float v, int n) {
    int i = blockIdx.x * blockDim.x + threadIdx.x;
    if (i < n) p[i] = v;
}

__global__ void deg_kernel(const int* __restrict__ dst, float* __restrict__ deg) {
    int e = blockIdx.x * blockDim.x + threadIdx.x;
    if (e < NE) atomicAdd(&deg[dst[e]], 1.0f);
}

__global__ void dinv_kernel(const float* __restrict__ deg, float* __restrict__ dinv) {
    int i = blockIdx.x * blockDim.x + threadIdx.x;
    if (i < NN) dinv[i] = rsqrtf(deg[i]);
}

// -------------------------------------------------- fp32 WMMA GEMM (16x16x4)
// C[M,N] = A[M,K] @ B[K,N] (+bias, +relu), row-major with strides.
// Block = 256 threads = 8 waves; block tile = 128 rows x 64 cols.
// B slab is staged through LDS (transposed, padded) in 128-row K-chunks and
// shared by all 8 waves; the k-loop is software-pipelined: loads for step i+1
// are issued before the WMMAs of step i.
template <int K, bool RELU>
__global__ __launch_bounds__(256) void gemm_wmma_f32(
    const float* __restrict__ A, const float* __restrict__ B,
    const float* __restrict__ bias, float* __restrict__ C,
    int M, int lda, int ldb, int ldc) {
    constexpr int KC  = 128;       // K-chunk held in LDS
    constexpr int LDK = KC + 2;    // pad: conflict-free transposed store, 8B align
    static_assert(K % KC == 0, "K must be a multiple of 128");
    __shared__ __align__(16) float Bs[64 * LDK];

    const int lane = threadIdx.x & 31;
    const int wave = threadIdx.x >> 5;
    const int r    = lane & 15;
    const int half = lane >> 4;                 // K split within fragment
    const int mtr  = blockIdx.x * 8 + wave;     // 16-row tile index
    const bool valid = (mtr * 16 < M);
    const int mt   = valid ? mtr : (M / 16 - 1);
    const int nb   = blockIdx.y * 64;           // first output column

    const float* __restrict__ Arow = A + (size_t)(mt * 16 + r) * lda + half * 2;

    const int lc = threadIdx.x & 63;            // B-load: column within slab
    const int lk = threadIdx.x >> 6;            // B-load: k row (stride 4)

    v8f acc[4];
#pragma unroll
    for (int t = 0; t < 4; ++t) acc[t] = (v8f){0.f,0.f,0.f,0.f,0.f,0.f,0.f,0.f};

    for (int kc = 0; kc < K; kc += KC) {
        __syncthreads();   // previous chunk fully consumed before overwrite
        // coalesced load of B[kc..kc+127][nb..nb+63], stored transposed
#pragma unroll
        for (int it = 0; it < KC / 4; ++it) {
            int k = lk + it * 4;
            Bs[lc * LDK + k] = B[(size_t)(kc + k) * ldb + nb + lc];
        }
        __syncthreads();

        // pipelined k-loop over the chunk
        v2f a_cur = *(const v2f*)(Arow + kc);
        v2f b_cur[4];
#pragma unroll
        for (int t = 0; t < 4; ++t)
            b_cur[t] = *(const v2f*)(&Bs[(t * 16 + r) * LDK + half * 2]);

#pragma unroll 4
        for (int k0 = 0; k0 < KC; k0 += 4) {
            v2f a_nxt = a_cur;
            v2f b_nxt[4];
            const int kn = k0 + 4;
            if (kn < KC) {
                a_nxt = *(const v2f*)(Arow + kc + kn);
#pragma unroll
                for (int t = 0; t < 4; ++t)
                    b_nxt[t] = *(const v2f*)(&Bs[(t * 16 + r) * LDK + kn + half * 2]);
            } else {
#pragma unroll
                for (int t = 0; t < 4; ++t) b_nxt[t] = b_cur[t];
            }
#pragma unroll
            for (int t = 0; t < 4; ++t)
                acc[t] = __builtin_amdgcn_wmma_f32_16x16x4_f32(
                    false, a_cur, false, b_cur[t], (short)0, acc[t], false, false);
            a_cur = a_nxt;
#pragma unroll
            for (int t = 0; t < 4; ++t) b_cur[t] = b_nxt[t];
        }
    }

    if (valid) {
#pragma unroll
        for (int t = 0; t < 4; ++t) {
            const int col = nb + t * 16 + r;
            const float bv = bias ? bias[col] : 0.0f;
#pragma unroll
            for (int v = 0; v < 8; ++v) {
                const int row = mt * 16 + half * 8 + v;
                float val = acc[t][v] + bv;
                if (RELU) val = fmaxf(val, 0.0f);
                C[(size_t)row * ldc + col] = val;
            }
        }
    }
}

// ------------------------------------------------------------- GCN pieces
// agg[i,f] = dinv[i]^2 * hw[i,f] + b[f]   (self-loop term + bias)
__global__ void selfloop_kernel(const float* __restrict__ hw,
                                const float* __restrict__ dinv,
                                const float* __restrict__ b,
                                float* __restrict__ agg) {
    int i = blockIdx.x * blockDim.x + threadIdx.x;
    if (i >= NN * HD) return;
    int node = i >> 7, f = i & 127;
    float di = dinv[node];
    agg[i] = di * di * hw[i] + b[f];
}

// 32 lanes per edge, 4 features per lane (float4 gather + 4 f32 atomics)
__global__ void edge_kernel(const int* __restrict__ src,
                            const int* __restrict__ dst,
                            const float* __restrict__ dinv,
                            const float* __restrict__ hw,
                            float* __restrict__ agg) {
    int t = blockIdx.x * blockDim.x + threadIdx.x;
    int e = t >> 5;
    if (e >= NE) return;
    int q = t & 31;
    int s = src[e], d = dst[e];
    float norm = dinv[s] * dinv[d];
    const float4* hs = (const float4*)(hw + (size_t)s * HD);
    float4 m = hs[q];
    float* ad = agg + (size_t)d * HD + q * 4;
    atomicAdd(ad + 0, norm * m.x);
    atomicAdd(ad + 1, norm * m.y);
    atomicAdd(ad + 2, norm * m.z);
    atomicAdd(ad + 3, norm * m.w);
}

// per-column sum / sumsq partials (128 threads = 1 column each, coalesced rows)
__global__ __launch_bounds__(128) void colstats_kernel(
    const float* __restrict__ h, float* __restrict__ sum, float* __restrict__ sq) {
    int f = threadIdx.x;
    int r0 = blockIdx.x * 256;
    int r1 = min(r0 + 256, NN);
    float s = 0.f, ss = 0.f;
    for (int r = r0; r < r1; ++r) {
        float v = h[(size_t)r * HD + f];
        s += v;
        ss += v * v;
    }
    atomicAdd(&sum[f], s);
    atomicAdd(&sq[f], ss);
}

// BN + ReLU, writing into the JK concat buffer (row stride ldo = NL*HD)
__global__ void bnrelu_kernel(const float* __restrict__ agg,
                              const float* __restrict__ sum,
                              const float* __restrict__ sq,
                              const float* __restrict__ gamma,
                              const float* __restrict__ beta,
                              float* __restrict__ out, int ldo) {
    int i = blockIdx.x * blockDim.x + threadIdx.x;
    if (i >= NN * HD) return;
    int node = i >> 7, f = i & 127;
    const float invn = 1.0f / (float)NN;
    float mu = sum[f] * invn;
    float var = sq[f] * invn - mu * mu;
    float v = (agg[i] - mu) * rsqrtf(var + BN_EPS) * gamma[f] + beta[f];
    out[(size_t)node * ldo + f] = fmaxf(v, 0.0f);
}

// -------------------------------------------------------------- pooling
__global__ void pool_cnt_kernel(const int* __restrict__ batch, float* __restrict__ cnt) {
    int i = blockIdx.x * blockDim.x + threadIdx.x;
    if (i < NN) atomicAdd(&cnt[batch[i]], 1.0f);
}

__global__ void pool_sum_kernel(const float* __restrict__ enc,
                                const int* __restrict__ batch,
                                float* __restrict__ gsum) {
    int t = blockIdx.x * blockDim.x + threadIdx.x;
    int i = t >> 6;                    // 64 quads of 4 floats = 256 features
    if (i >= NN) return;
    int q = t & 63;
    int g = batch[i];
    const float4* er = (const float4*)(enc + (size_t)i * ENCD);
    float4 m = er[q];
    float* gs = gsum + (size_t)g * ENCD + q * 4;
    atomicAdd(gs + 0, m.x);
    atomicAdd(gs + 1, m.y);
    atomicAdd(gs + 2, m.z);
    atomicAdd(gs + 3, m.w);
}

__global__ void pool_div_kernel(float* __restrict__ gsum, const float* __restrict__ cnt) {
    int i = blockIdx.x * blockDim.x + threadIdx.x;
    if (i >= NG * ENCD) return;
    int g = i >> 8;
    gsum[i] /= fmaxf(cnt[g], 1.0f);
}

// -------------------------------------------------------------- decoder
__global__ __launch_bounds__(128) void dec1_kernel(const float* __restrict__ pooled,
                                                   const float* __restrict__ W1,
                                                   const float* __restrict__ b1,
                                                   float* __restrict__ hidden) {
    __shared__ float p[ENCD];
    int g = blockIdx.x;
    for (int i = threadIdx.x; i < ENCD; i += 128) p[i] = pooled[(size_t)g * ENCD + i];
    __syncthreads();
    int j = threadIdx.x;
    float acc = b1[j];
    for (int k = 0; k < ENCD; ++k) acc += p[k] * W1[(size_t)k * DECD + j];
    hidden[(size_t)g * DECD + j] = fmaxf(acc, 0.0f);
}

__global__ __launch_bounds__(128) void dec2_kernel(const float* __restrict__ hidden,
                                                   const float* __restrict__ W2,
                                                   const float* __restrict__ b2,
                                                   float* __restrict__ out) {
    __shared__ float red[128];
    int g = blockIdx.x;
    int j = threadIdx.x;
    red[j] = hidden[(size_t)g * DECD + j] * W2[j];
    __syncthreads();
    for (int s = 64; s > 0; s >>= 1) {
        if (j < s) red[j] += red[j + s];
        __syncthreads();
    }
    if (j == 0) out[g] = red[0] + b2[0];
}

// ---------------------------------------------------------------- driver
extern "C" void kernel_launch(void* const* d_in, const int* in_sizes, int n_in,
                              void* d_out, int out_size, void* d_ws, size_t ws_size,
                              hipStream_t stream) {
    const float* x        = (const float*)d_in[0];
    const int*   ei       = (const int*)d_in[1];
    const int*   batch    = (const int*)d_in[2];
    const float* conv_W   = (const float*)d_in[3];   // [L,H,H]
    const float* conv_b   = (const float*)d_in[4];   // [L,H]
    const float* bn_gamma = (const float*)d_in[5];   // [L,H]
    const float* bn_beta  = (const float*)d_in[6];   // [L,H]
    const float* enc_W    = (const float*)d_in[7];   // [512,256]
    const float* enc_b    = (const float*)d_in[8];   // [256]
    const float* dec_W1   = (const float*)d_in[9];   // [256,128]
    const float* dec_b1   = (const float*)d_in[10];  // [128]
    const float* dec_W2   = (const float*)d_in[11];  // [128,1]
    const float* dec_b2   = (const float*)d_in[12];  // [1]
    float* out = (float*)d_out;

    // workspace carve-out (256B aligned blocks)
    char* ws = (char*)d_ws;
    size_t off = 0;
    auto carve = [&](size_t nfl) {
        float* p = (float*)(ws + off);
        off += ((nfl * sizeof(float) + 255) / 256) * 256;
        return p;
    };
    float* deg    = carve(NN);
    float* dinv   = carve(NN);
    float* hw     = carve((size_t)NN * HD);
    float* agg    = carve((size_t)NN * HD);
    float* hk     = carve((size_t)NN * NL * HD);   // JK concat [N, 512]
    float* enc    = carve((size_t)NN * ENCD);
    float* colsum = carve(HD);
    float* colsq  = carve(HD);
    float* gsum   = carve((size_t)NG * ENCD);
    float* gcnt   = carve(NG);
    float* hidden = carve((size_t)NG * DECD);

    const int* src = ei;
    const int* dst = ei + NE;

    const int B = 256;
    auto blocks = [](long long n, int b) { return (unsigned)((n + b - 1) / b); };

    // degrees with self-loop, dinv
    fill_kernel<<<blocks(NN, B), B, 0, stream>>>(deg, 1.0f, NN);
    deg_kernel<<<blocks(NE, B), B, 0, stream>>>(dst, deg);
    dinv_kernel<<<blocks(NN, B), B, 0, stream>>>(deg, dinv);

    const unsigned mblocks = (NN / 16 + 7) / 8;  // 6250 tiles -> 782 blocks

    for (int l = 0; l < NL; ++l) {
        const float* hin = (l == 0) ? x : (hk + (size_t)(l - 1) * HD);
        const int lda    = (l == 0) ? HD : NL * HD;
        // hw = h @ W_l  (fp32 WMMA, LDS-staged B)
        gemm_wmma_f32<HD, false><<<dim3(mblocks, HD / 64), 256, 0, stream>>>(
            hin, conv_W + (size_t)l * HD * HD, nullptr, hw, NN, lda, HD, HD);
        // agg = dinv^2 * hw + b   then scatter-add edge messages
        selfloop_kernel<<<blocks((long long)NN * HD, B), B, 0, stream>>>(
            hw, dinv, conv_b + (size_t)l * HD, agg);
        edge_kernel<<<blocks((long long)NE * 32, B), B, 0, stream>>>(
            src, dst, dinv, hw, agg);
        // BN stats + BN/ReLU into JK slot l
        fill_kernel<<<1, 128, 0, stream>>>(colsum, 0.0f, HD);
        fill_kernel<<<1, 128, 0, stream>>>(colsq, 0.0f, HD);
        colstats_kernel<<<blocks(NN, 256), 128, 0, stream>>>(agg, colsum, colsq);
        bnrelu_kernel<<<blocks((long long)NN * HD, B), B, 0, stream>>>(
            agg, colsum, colsq, bn_gamma + (size_t)l * HD, bn_beta + (size_t)l * HD,
            hk + (size_t)l * HD, NL * HD);
    }

    // encoder: enc = relu(hk @ enc_W + enc_b)   [100000,512]@[512,256]
    gemm_wmma_f32<NL * HD, true><<<dim3(mblocks, ENCD / 64), 256, 0, stream>>>(
        hk, enc_W, enc_b, enc, NN, NL * HD, ENCD, ENCD);

    // scatter-mean pooling over batch
    fill_kernel<<<blocks(NG * ENCD, B), B, 0, stream>>>(gsum, 0.0f, NG * ENCD);
    fill_kernel<<<1, NG, 0, stream>>>(gcnt, 0.0f, NG);
    pool_cnt_kernel<<<blocks(NN, B), B, 0, stream>>>(batch, gcnt);
    pool_sum_kernel<<<blocks((long long)NN * 64, B), B, 0, stream>>>(enc, batch, gsum);
    pool_div_kernel<<<blocks(NG * ENCD, B), B, 0, stream>>>(gsum, gcnt);

    // decoder MLP
    dec1_kernel<<<NG, 128, 0, stream>>>(gsum, dec_W1, dec_b1, hidden);
    dec2_kernel<<<NG, 128, 0, stream>>>(hidden, dec_W2, dec_b2, out);
}